// SparseGraphWaveletLayer_10316511445513
// MI455X (gfx1250) — compile-verified
//
#include <hip/hip_runtime.h>
#include <hip/hip_bf16.h>

// ---------------------------------------------------------------------------
// SparseGraphWaveletLayer for MI455X (gfx1250, wave32).
//
// Strategy (roofline): all dense matrices (filtered/z: 25.6MB, out: 76.8MB,
// dense_feat: 51.2MB) fit in the 192MB L2 -> scatter-atomic COO SpMM runs at
// L2 atomic throughput, not HBM. Index/value streams (~134MB total) are read
// once with non-temporal loads to avoid evicting the hot working set.
// The feature x weight product is densified and run through the f32 matrix
// pipe with V_WMMA_F32_16X16X4_F32 (16x16 tile per wave, K=128 in 32 steps).
// ---------------------------------------------------------------------------

typedef __attribute__((ext_vector_type(2))) float v2f;
typedef __attribute__((ext_vector_type(8))) float v8f;

#define IN_C 128
#define OUT_C 64

// --- Step 1a: densify sparse feature matrix: dense[r*128+c] += v -----------
__global__ __launch_bounds__(256) void feat_scatter_kernel(
    const int* __restrict__ rows, const int* __restrict__ cols,
    const float* __restrict__ vals, float* __restrict__ dense, int nnz) {
  int k = blockIdx.x * blockDim.x + threadIdx.x;
  if (k < nnz) {
    int r = __builtin_nontemporal_load(rows + k);
    int c = __builtin_nontemporal_load(cols + k);
    float v = __builtin_nontemporal_load(vals + k);
    atomicAdd(&dense[(long long)r * IN_C + c], v);
  }
}

// --- Step 1b: filtered[N,64] = dense[N,128] x W[128,64] via f32 WMMA -------
// One block = one 16-row M tile; 4 waves cover the 4 N tiles (64 cols).
// A 16x4 f32 operand layout: lane m=lane&15; lanes>=16 hold K+2; float2 = K,K+1.
// C/D layout: 8 VGPRs, row = vgpr + 8*(lane>=16), col = lane&15.
__global__ __launch_bounds__(128) void gemm_wmma_kernel(
    const float* __restrict__ A, const float* __restrict__ W,
    float* __restrict__ C) {
  const int wave = threadIdx.x >> 5;
  const int lane = threadIdx.x & 31;
  const int m0 = blockIdx.x * 16;
  const int n0 = wave * 16;
  const int mRow = m0 + (lane & 15);
  const int nCol = n0 + (lane & 15);
  const int kHalf = (lane >> 4) * 2;  // lanes 16..31 handle K+2, K+3

  v8f acc = {};
  const float* aRow = A + (long long)mRow * IN_C;
#pragma unroll 4
  for (int k = 0; k < IN_C; k += 4) {
    v2f a = *(const v2f*)(aRow + k + kHalf);          // A[m][k+kHalf .. +1]
    v2f b;
    b.x = W[(k + kHalf) * OUT_C + nCol];              // B[k+kHalf][n]
    b.y = W[(k + kHalf + 1) * OUT_C + nCol];          // B[k+kHalf+1][n]
    acc = __builtin_amdgcn_wmma_f32_16x16x4_f32(
        false, a, false, b, (short)0, acc, false, false);
  }

  const int rBase = m0 + ((lane >> 4) << 3);
#pragma unroll
  for (int v = 0; v < 8; ++v) {
    C[(long long)(rBase + v) * OUT_C + nCol] = acc[v];
  }
}

// --- COO SpMM scatter: dst[r*stride+off+ch] += v*(theta?theta[c]:1)*src[c*64+ch]
// 64 threads = one output row's channels; each block loops NZ_PER_BLOCK nnz.
// Index/value streams are one-shot -> non-temporal loads keep L2 for the
// dense src/dst buffers. src row read is a fully coalesced 256B burst.
#define NZ_PER_BLOCK 16
__global__ __launch_bounds__(64) void spmm_scatter_kernel(
    const int* __restrict__ rows, const int* __restrict__ cols,
    const float* __restrict__ vals, const float* __restrict__ theta,
    const float* __restrict__ src, float* __restrict__ dst,
    int nnz, int dstStride, int dstOff) {
  const int ch = threadIdx.x;
  int k = blockIdx.x * NZ_PER_BLOCK;
  int kEnd = k + NZ_PER_BLOCK;
  if (kEnd > nnz) kEnd = nnz;
  for (; k < kEnd; ++k) {
    int r = __builtin_nontemporal_load(rows + k);
    int c = __builtin_nontemporal_load(cols + k);
    float v = __builtin_nontemporal_load(vals + k);
    if (theta) v *= theta[c];
    float x = v * src[(long long)c * OUT_C + ch];
    atomicAdd(&dst[(long long)r * dstStride + dstOff + ch], x);
  }
}

// --- Final ReLU in place on d_out ------------------------------------------
__global__ __launch_bounds__(256) void relu_kernel(float* __restrict__ x, int n) {
  int i = blockIdx.x * blockDim.x + threadIdx.x;
  if (i < n) x[i] = fmaxf(x[i], 0.0f);
}

extern "C" void kernel_launch(void* const* d_in, const int* in_sizes, int n_in,
                              void* d_out, int out_size, void* d_ws, size_t ws_size,
                              hipStream_t stream) {
  // Inputs (setup_inputs dict order):
  const int*   phi_idx   = (const int*)d_in[0];    // (S,2,NNZ) int32
  const float* phi_val   = (const float*)d_in[1];  // (S,NNZ)
  const int*   phii_idx  = (const int*)d_in[2];    // (S,2,NNZ)
  const float* phii_val  = (const float*)d_in[3];  // (S,NNZ)
  const int*   feat_idx  = (const int*)d_in[4];    // (2,NNZ_F)
  const float* feat_val  = (const float*)d_in[5];  // (NNZ_F,)
  const float* Wm        = (const float*)d_in[6];  // (128,64) row-major
  const float* theta     = (const float*)d_in[7];  // (N,1)
  float*       out       = (float*)d_out;          // (N, S, 64)

  const int N    = in_sizes[7];                    // 100000
  const int NNZF = in_sizes[5];                    // 1600000
  const int S    = out_size / (N * OUT_C);         // 3
  const int NNZ  = in_sizes[1] / S;                // 1600000

  // Workspace layout (76.8 MB total):
  //   [0, N*64)        filtered
  //   [N*64, N*192)    dense_feat (phase 1, N*128 floats) -> reused as z (N*64)
  float* filtered   = (float*)d_ws;
  float* dense_feat = filtered + (long long)N * OUT_C;
  float* z          = dense_feat;                  // alias: dense_feat dead by then

  // Zero accumulators (graph-capturable memset nodes).
  hipMemsetAsync(d_ws, 0, (size_t)N * (OUT_C + IN_C) * sizeof(float), stream);
  hipMemsetAsync(d_out, 0, (size_t)out_size * sizeof(float), stream);

  // 1) densify features, then dense f32-WMMA GEMM -> filtered
  feat_scatter_kernel<<<(NNZF + 255) / 256, 256, 0, stream>>>(
      feat_idx, feat_idx + NNZF, feat_val, dense_feat, NNZF);
  gemm_wmma_kernel<<<N / 16, 128, 0, stream>>>(dense_feat, Wm, filtered);

  // 2) per scale: z = phi_inv @ filtered ; out[:,i,:] += phi @ (theta*z)
  const int spmmGrid = (NNZ + NZ_PER_BLOCK - 1) / NZ_PER_BLOCK;
  for (int i = 0; i < S; ++i) {
    hipMemsetAsync(z, 0, (size_t)N * OUT_C * sizeof(float), stream);
    const int* ri = phii_idx + (long long)i * 2 * NNZ;
    spmm_scatter_kernel<<<spmmGrid, 64, 0, stream>>>(
        ri, ri + NNZ, phii_val + (long long)i * NNZ, nullptr,
        filtered, z, NNZ, OUT_C, 0);
    const int* r2 = phi_idx + (long long)i * 2 * NNZ;
    spmm_scatter_kernel<<<spmmGrid, 64, 0, stream>>>(
        r2, r2 + NNZ, phi_val + (long long)i * NNZ, theta,
        z, out, NNZ, S * OUT_C, i * OUT_C);
  }

  // 3) ReLU in place
  int total = N * S * OUT_C;
  relu_kernel<<<(total + 255) / 256, 256, 0, stream>>>(out, total);
}